// QAgent_26963804684672
// MI455X (gfx1250) — compile-verified
//
#include <hip/hip_runtime.h>
#include <math.h>

// ---------------------------------------------------------------------------
// QAgent selective-scan for MI455X (gfx1250, wave32, WMMA + TDM).
//
// Dims (fixed by reference): T=2048, BK=64, OBS=64, H=16, P=64, N=64, DIN=1024
//
// Design: sequential-scan latency bound -> 1024 independent (b,h) scans with
// register-resident state (32 f32/thread, 128 thr/WG); u-projection fused per
// 16-step block via V_WMMA_F32_16X16X4_F32 with W_in fragments pinned in
// VGPRs; B/C projections precomputed with WMMA into workspace (64MB, L2
// resident); all tile staging done with Tensor Data Mover DMA
// (tensor_load_to_lds) double-buffered across 16-step blocks so global/L2
// latency is hidden behind the WMMA + scan critical path.
// ---------------------------------------------------------------------------

#define T_DIM   2048
#define BK_DIM  64
#define OBS_DIM 64
#define H_DIM   16
#define P_DIM   64
#define N_DIM   64
#define DIN_DIM 1024

typedef float v2f __attribute__((ext_vector_type(2)));
typedef float v8f __attribute__((ext_vector_type(8)));
typedef unsigned int u32x4 __attribute__((ext_vector_type(4)));
typedef int i32x4 __attribute__((ext_vector_type(4)));
typedef int i32x8 __attribute__((ext_vector_type(8)));

// Low 32 bits of a flat shared-aperture pointer == LDS byte address (ISA 10.2)
#define LDS_ADDR(p) ((unsigned)(unsigned long long)(size_t)(p))

// ---------------------------------------------------------------------------
// TDM 2-D tile load: tile_h rows x tile_w f32 elements, global row stride
// row_stride_elems, optional LDS padding (pad codes per ISA D# group1).
// Descriptor packing per CDNA5 ISA §8.3/8.4 (groups 0/1; 2D -> groups 2/3 zero).
// Issue from ONE wave only (EXEC ignored; wave-level op, TENSORcnt tracked).
// Toolchain note: 6-arg builtin form (clang-23/therock-10.0).
// ---------------------------------------------------------------------------
__device__ __forceinline__ void tdm_load_2d(unsigned lds_addr,
                                            const float* gptr,
                                            unsigned tile_w, unsigned tile_h,
                                            unsigned row_stride_elems,
                                            unsigned pad_en,
                                            unsigned pad_interval,
                                            unsigned pad_amount) {
    unsigned long long ga = (unsigned long long)(size_t)gptr;
    u32x4 g0;
    g0.x = 1u;                                           // count=1 (valid D#)
    g0.y = lds_addr;                                     // lds_addr [63:32]
    g0.z = (unsigned)ga;                                 // global_addr lo
    g0.w = (unsigned)((ga >> 32) & 0x01FFFFFFull)        // global_addr [56:32]
         | (2u << 30);                                   // type=2 ("image")
    i32x8 g1;
    g1[0] = (int)((2u << 16)                             // data_size=2 -> 4B
                | (pad_en << 20)
                | (pad_interval << 22)
                | (pad_amount << 25));
    g1[1] = (int)((tile_w & 0xFFFFu) << 16);             // tensor_dim0[15:0]
    g1[2] = (int)((tile_h & 0xFFFFu) << 16);             // tensor_dim1[15:0]
    g1[3] = (int)((tile_w & 0xFFFFu) << 16);             // tile_dim0
    g1[4] = (int)(tile_h & 0xFFFFu);                     // tile_dim1
    g1[5] = (int)row_stride_elems;                       // tensor_dim0_stride
    g1[6] = 0;
    g1[7] = 0;
    i32x4 z4 = {0, 0, 0, 0};
    i32x8 z8 = {0, 0, 0, 0, 0, 0, 0, 0};
    __builtin_amdgcn_tensor_load_to_lds(g0, g1, z4, z4, z8, 0);
}

// ---------------------------------------------------------------------------
// Kernel 1: dt = softplus(reward * W_dt + dt_bias); decay = exp(dt * -exp(A_log))
// Layout: [t][b][h]
// ---------------------------------------------------------------------------
__global__ void dt_decay_kernel(const float* __restrict__ reward,
                                const float* __restrict__ W_dt,
                                const float* __restrict__ dt_bias,
                                const float* __restrict__ A_log,
                                float* __restrict__ dt_out,
                                float* __restrict__ dec_out) {
    int idx = blockIdx.x * blockDim.x + threadIdx.x;   // < T*BK*H
    int h  = idx & (H_DIM - 1);
    int tb = idx >> 4;                                 // t*BK + b
    float r = reward[tb];
    float x = fmaf(r, W_dt[h], dt_bias[h]);
    float sp = (x > 20.0f) ? x : log1pf(expf(x));      // stable softplus
    float A  = -expf(A_log[h]);
    dt_out[idx]  = sp;
    dec_out[idx] = expf(sp * A);
}

// ---------------------------------------------------------------------------
// Kernel 2: Bm = obs @ W_B, Cm = obs @ W_C   (M=T*BK=131072, K=64, N=64)
// One WG (256 thr = 8 waves) per 16-row M-tile; obs tile staged by TDM.
// Waves 0-3 -> B col-tiles 0..3, waves 4-7 -> C col-tiles 0..3.
// ---------------------------------------------------------------------------
__global__ void proj_bc_kernel(const float* __restrict__ obs,
                               const float* __restrict__ W_B,
                               const float* __restrict__ W_C,
                               float* __restrict__ Bm,
                               float* __restrict__ Cm) {
    __shared__ float obs_s[16][68];     // padded stride (TDM pad: 64dw + 4dw)

    const int tid  = threadIdx.x;
    const int mt   = blockIdx.x;        // M-tile (16 rows of T*BK)
    const int lane = tid & 31;
    const int wave = tid >> 5;

    // stage obs tile with one TDM DMA (rows contiguous: stride = 64 elems)
    if (wave == 0)
        tdm_load_2d(LDS_ADDR(&obs_s[0][0]), obs + (size_t)mt * 16 * OBS_DIM,
                    64, 16, 64, /*pad*/1, /*interval 64dw*/5, /*amount 4dw*/3);
    __builtin_amdgcn_s_wait_tensorcnt(0);
    __syncthreads();

    const float* W   = (wave < 4) ? W_B : W_C;
    float*       Out = (wave < 4) ? Bm  : Cm;
    const int nt   = wave & 3;                  // 16-col N tile
    const int mrow = lane & 15;
    const int koff = (lane >> 4) << 1;          // 0 for lanes 0-15, 2 for 16-31
    const int ncol = nt * 16 + (lane & 15);

    v8f acc = {};
    #pragma unroll
    for (int kk = 0; kk < 16; ++kk) {
        const int k0 = kk * 4;
        v2f a, bf;
        a.x  = obs_s[mrow][k0 + koff];
        a.y  = obs_s[mrow][k0 + koff + 1];
        bf.x = W[(k0 + koff)     * N_DIM + ncol];
        bf.y = W[(k0 + koff + 1) * N_DIM + ncol];
        acc = __builtin_amdgcn_wmma_f32_16x16x4_f32(false, a, false, bf,
                                                    (short)0, acc, false, false);
    }
    // D layout: VGPR r -> M=r (lanes 0-15) / M=r+8 (lanes 16-31)
    const int mbase = (lane >> 4) << 3;
    #pragma unroll
    for (int r = 0; r < 8; ++r)
        Out[(mt * 16 + mbase + r) * N_DIM + ncol] = acc[r];
}

// ---------------------------------------------------------------------------
// Kernel 3: fused u-projection (WMMA) + selective scan, TDM double-buffered.
// Grid = 1024 WGs, one per (b,h).  Block = 128 (4 waves).
// State: P x N = 4096 f32 per WG -> 32 registers/thread.
//   thread owns p = tid&63, n-range = [ (tid>>6)*32, +32 )
// Per 16-step block: wave0 TDM-prefetches next block's obs/B/C tiles into the
// alternate LDS buffer while this block runs WMMA u-projection + 16 scan steps.
// ---------------------------------------------------------------------------
__global__ void scan_kernel(const float* __restrict__ obs,
                            const float* __restrict__ W_in,
                            const float* __restrict__ W_out,
                            const float* __restrict__ Dvec,
                            const float* __restrict__ init_state,
                            const float* __restrict__ Bm,
                            const float* __restrict__ Cm,
                            const float* __restrict__ dt_ws,
                            const float* __restrict__ dec_ws,
                            float* __restrict__ partial) {
    __shared__ float obs_s[2][16][68];      // TDM-padded (stride 68)
    __shared__ float Bs[2][16][64];
    __shared__ float Cs[2][16][64];
    __shared__ float u_s[16][68];
    __shared__ float dts[2][16];
    __shared__ float decs[2][16];
    __shared__ float red[16][4];

    const int tid  = threadIdx.x;
    const int lane = tid & 31;
    const int wave = tid >> 5;
    const int b    = blockIdx.x & (BK_DIM - 1);
    const int h    = blockIdx.x >> 6;

    const int p  = tid & 63;        // owned P row
    const int ng = tid >> 6;        // n half: 0 or 1
    const int n0 = ng * 32;

    const float wout = W_out[h * P_DIM + p];
    const float Dh   = Dvec[h];

    // register-resident state: h0 = init_state[h,p,n0..n0+31] (broadcast over b)
    float st[32];
    #pragma unroll
    for (int j = 0; j < 32; ++j)
        st[j] = init_state[(h * P_DIM + p) * N_DIM + n0 + j];

    // Pin W_in B-fragments in VGPRs: wave w computes u cols [w*16, w*16+16).
    const int pc   = wave * 16 + (lane & 15);
    const int koff = (lane >> 4) << 1;
    v2f bw[16];
    #pragma unroll
    for (int kk = 0; kk < 16; ++kk) {
        const int k0 = kk * 4;
        bw[kk].x = W_in[(k0 + koff)     * DIN_DIM + h * P_DIM + pc];
        bw[kk].y = W_in[(k0 + koff + 1) * DIN_DIM + h * P_DIM + pc];
    }

    const int mrow = lane & 15;
    // global tile bases for this b: row (t) stride = BK*64 elements
    const float* obs_b = obs + (size_t)b * OBS_DIM;
    const float* Bm_b  = Bm  + (size_t)b * N_DIM;
    const float* Cm_b  = Cm  + (size_t)b * N_DIM;

    // ---- prologue: stage block 0 into buffer 0 ----
    if (wave == 0) {
        tdm_load_2d(LDS_ADDR(&obs_s[0][0][0]), obs_b, 64, 16, BK_DIM * OBS_DIM,
                    1, 5, 3);
        tdm_load_2d(LDS_ADDR(&Bs[0][0][0]), Bm_b, 64, 16, BK_DIM * N_DIM, 0, 0, 0);
        tdm_load_2d(LDS_ADDR(&Cs[0][0][0]), Cm_b, 64, 16, BK_DIM * N_DIM, 0, 0, 0);
    }
    if (tid < 16) {
        const int idx = (tid * BK_DIM + b) * H_DIM + h;
        dts[0][tid]  = dt_ws[idx];
        decs[0][tid] = dec_ws[idx];
    }
    __builtin_amdgcn_s_wait_tensorcnt(0);
    __syncthreads();

    int buf = 0;
    for (int tb = 0; tb < T_DIM / 16; ++tb) {
        const int t0 = tb * 16;
        const int nb = buf ^ 1;

        // ---- async prefetch of next block into alternate buffer ----
        if (tb + 1 < T_DIM / 16) {
            const int t0n = t0 + 16;
            if (wave == 0) {
                tdm_load_2d(LDS_ADDR(&obs_s[nb][0][0]),
                            obs_b + (size_t)t0n * BK_DIM * OBS_DIM,
                            64, 16, BK_DIM * OBS_DIM, 1, 5, 3);
                tdm_load_2d(LDS_ADDR(&Bs[nb][0][0]),
                            Bm_b + (size_t)t0n * BK_DIM * N_DIM,
                            64, 16, BK_DIM * N_DIM, 0, 0, 0);
                tdm_load_2d(LDS_ADDR(&Cs[nb][0][0]),
                            Cm_b + (size_t)t0n * BK_DIM * N_DIM,
                            64, 16, BK_DIM * N_DIM, 0, 0, 0);
            }
            if (tid >= 32 && tid < 48) {
                const int i = tid - 32;
                dts[nb][i] = dt_ws[((t0n + i) * BK_DIM + b) * H_DIM + h];
            } else if (tid >= 48 && tid < 64) {
                const int i = tid - 48;
                decs[nb][i] = dec_ws[((t0n + i) * BK_DIM + b) * H_DIM + h];
            }
        }

        // ---- u GEMM: 16x64 = 16x64 @ 64x64 (per-head W_in slice) ----
        v8f acc = {};
        #pragma unroll
        for (int kk = 0; kk < 16; ++kk) {
            const int k0 = kk * 4;
            v2f a;
            a.x = obs_s[buf][mrow][k0 + koff];
            a.y = obs_s[buf][mrow][k0 + koff + 1];
            acc = __builtin_amdgcn_wmma_f32_16x16x4_f32(false, a, false, bw[kk],
                                                        (short)0, acc, false, false);
        }
        {
            const int col   = wave * 16 + (lane & 15);
            const int mbase = (lane >> 4) << 3;
            #pragma unroll
            for (int r = 0; r < 8; ++r)
                u_s[mbase + r][col] = acc[r];
        }
        __syncthreads();

        // ---- 16 sequential scan steps ----
        #pragma unroll 1
        for (int s = 0; s < 16; ++s) {
            const float dtc = dts[buf][s];
            const float dc  = decs[buf][s];
            const float uu  = u_s[s][p];
            const float du  = dtc * uu;
            float yp = 0.0f;
            #pragma unroll
            for (int j = 0; j < 32; ++j) {
                const float bv = Bs[buf][s][n0 + j];   // wave-uniform broadcast
                st[j] = fmaf(st[j], dc, du * bv);
                yp    = fmaf(st[j], Cs[buf][s][n0 + j], yp);
            }
            float contrib = yp * wout;
            if (ng == 0) contrib = fmaf(Dh * uu, wout, contrib);
            #pragma unroll
            for (int off = 16; off >= 1; off >>= 1)
                contrib += __shfl_xor(contrib, off, 32);
            if (lane == 0) red[s][wave] = contrib;
        }
        __syncthreads();
        if (tid < 16) {
            const float o4 = red[tid][0] + red[tid][1] + red[tid][2] + red[tid][3];
            partial[((t0 + tid) * BK_DIM + b) * H_DIM + h] = o4;
        }
        // next buffer must be fully DMA'd before flipping
        __builtin_amdgcn_s_wait_tensorcnt(0);
        __syncthreads();
        buf = nb;
    }
}

// ---------------------------------------------------------------------------
// Kernel 4: q[t,b] = sum_h partial[t,b,h]; write tuple (q, logits) = (q, q)
// ---------------------------------------------------------------------------
__global__ void reduce_kernel(const float* __restrict__ partial,
                              float* __restrict__ out,
                              int write_second) {
    const int i = blockIdx.x * blockDim.x + threadIdx.x;   // < T*BK
    float s = 0.0f;
    #pragma unroll
    for (int hh = 0; hh < H_DIM; ++hh)
        s += partial[i * H_DIM + hh];
    out[i] = s;
    if (write_second)
        out[T_DIM * BK_DIM + i] = s;
}

// ---------------------------------------------------------------------------
// Launch
// ---------------------------------------------------------------------------
extern "C" void kernel_launch(void* const* d_in, const int* in_sizes, int n_in,
                              void* d_out, int out_size, void* d_ws, size_t ws_size,
                              hipStream_t stream) {
    const float* obs        = (const float*)d_in[0];
    const float* reward     = (const float*)d_in[1];
    const float* W_in       = (const float*)d_in[2];
    const float* W_B        = (const float*)d_in[3];
    const float* W_C        = (const float*)d_in[4];
    const float* W_dt       = (const float*)d_in[5];
    const float* dt_bias    = (const float*)d_in[6];
    const float* A_log      = (const float*)d_in[7];
    const float* Dvec       = (const float*)d_in[8];
    const float* W_out      = (const float*)d_in[9];
    const float* init_state = (const float*)d_in[10];
    (void)in_sizes; (void)n_in; (void)ws_size;

    // workspace layout (floats):
    //   Bm: T*BK*N (8.4M) | Cm: 8.4M | dt: T*BK*H (2.1M) | dec: 2.1M | partial: 2.1M
    float* ws      = (float*)d_ws;
    float* Bm      = ws;
    float* Cm      = Bm + (size_t)T_DIM * BK_DIM * N_DIM;
    float* dt_ws   = Cm + (size_t)T_DIM * BK_DIM * N_DIM;
    float* dec_ws  = dt_ws + (size_t)T_DIM * BK_DIM * H_DIM;
    float* partial = dec_ws + (size_t)T_DIM * BK_DIM * H_DIM;

    // 1) dt / decay   (T*BK*H = 2,097,152 elems)
    dt_decay_kernel<<<dim3((T_DIM * BK_DIM * H_DIM) / 256), dim3(256), 0, stream>>>(
        reward, W_dt, dt_bias, A_log, dt_ws, dec_ws);

    // 2) B & C projections (WMMA, TDM-staged): 8192 M-tiles
    proj_bc_kernel<<<dim3((T_DIM * BK_DIM) / 16), dim3(256), 0, stream>>>(
        obs, W_B, W_C, Bm, Cm);

    // 3) fused WMMA u-projection + register-state scan: 1024 (b,h) WGs
    scan_kernel<<<dim3(BK_DIM * H_DIM), dim3(128), 0, stream>>>(
        obs, W_in, W_out, Dvec, init_state, Bm, Cm, dt_ws, dec_ws, partial);

    // 4) reduce over heads -> q (and logits == q)
    const int write_second = (out_size >= 2 * T_DIM * BK_DIM) ? 1 : 0;
    reduce_kernel<<<dim3((T_DIM * BK_DIM) / 256), dim3(256), 0, stream>>>(
        partial, (float*)d_out, write_second);
}